// ModelIAS_53618371724066
// MI455X (gfx1250) — compile-verified
//
#include <hip/hip_runtime.h>
#include <stdint.h>

typedef __attribute__((ext_vector_type(2))) float v2f;
typedef __attribute__((ext_vector_type(8))) float v8f;

#define BB 256
#define TT 256      // tokens after [CLS]
#define HH 768
#define SS 130
#define PA 44       // padded A-chunk row stride (floats): rows 16B-aligned, gcd(44,64)=4 -> 16 distinct banks
#define PL 133      // padded logits row stride (floats): odd -> conflict-free column walks
#define KC 32       // K chunk staged per iteration
#define NTHREADS 288

#define A_BYTES   (256 * PA * 4)                 // 45056 per buffer
#define L_BYTES   (256 * PL * 4)                 // 136192
// LDS: A-chunk double buffer | logits | seg | cnt | start
#define LDS_BYTES (2 * A_BYTES + L_BYTES + 3 * 256 * 4)   // 229376 < 320K WGP LDS

// Issue per-lane async global->LDS copies of one 256x32 fp32 A-chunk (ASYNCcnt tracked,
// no data VGPRs consumed; overlaps with WMMA compute on the other buffer).
__device__ __forceinline__ void stage_chunk_async(const float* __restrict__ hidA,
                                                  int kc, float* ldsbuf, int tid)
{
    for (int j = tid; j < (TT * KC) / 4; j += NTHREADS) {
        int row = j >> 3;
        int c4  = (j & 7) << 2;
        const float* gp = hidA + (size_t)row * HH + kc + c4;           // 16B aligned
        unsigned ldsoff = (unsigned)(uintptr_t)(&ldsbuf[row * PA + c4]); // LDS byte offset in addr[31:0]
        unsigned long long ga = (unsigned long long)(uintptr_t)gp;
        asm volatile("global_load_async_to_lds_b128 %0, %1, off"
                     :
                     : "v"(ldsoff), "v"(ga));
    }
}

__device__ __forceinline__ void wait_async_zero()
{
    asm volatile("s_wait_asynccnt 0x0" ::: "memory");
}

__global__ __launch_bounds__(NTHREADS, 1)
void slotmerge_kernel(const float* __restrict__ hidden,  // [B, T+1, H]
                      const float* __restrict__ W,       // [H, S]
                      const float* __restrict__ bias,    // [S]
                      const int*   __restrict__ seg,     // [B, T] sorted per row
                      float*       __restrict__ out)     // [B, S, T]
{
    extern __shared__ char smem[];
    float* ldsA0   = (float*)smem;                                 // [256][PA]
    float* ldsA1   = (float*)(smem + A_BYTES);                     // [256][PA]
    float* ldsL    = (float*)(smem + 2 * A_BYTES);                 // [256][PL]
    int*   ldsSeg  = (int*)  (smem + 2 * A_BYTES + L_BYTES);       // [256]
    int*   ldsCnt  = ldsSeg + 256;                                 // [256]
    int*   ldsStart= ldsCnt + 256;                                 // [256]

    const int tid  = threadIdx.x;
    const int b    = blockIdx.x;
    const int lane = tid & 31;
    const int wv   = tid >> 5;          // 0..8 : which 16-wide S tile this wave owns
    const int m    = lane & 15;
    const int hi   = lane >> 4;         // 0 or 1 (half-wave)
    const int hi2  = hi * 2;

    // ---- phase 0: segment bookkeeping (sorted runs -> count & start per word) ----
    if (tid < 256) {
        ldsSeg[tid] = seg[b * TT + tid];
        ldsCnt[tid] = 0;
    }
    __syncthreads();
    if (tid < 256) {
        int w = ldsSeg[tid];
        atomicAdd(&ldsCnt[w], 1);
        if (tid == 0 || ldsSeg[tid - 1] != w) ldsStart[w] = tid;
    }

    // ---- phase 1: logits = hidden[b,1:,:] @ W + bias, via V_WMMA_F32_16X16X4_F32 ----
    const int   sc  = wv * 16 + m;                 // slot column this lane produces
    const int   scc = (sc < SS) ? sc : (SS - 1);   // clamped for safe (garbage) pad loads
    const float bv  = (sc < SS) ? bias[sc] : 0.0f;

    v8f acc[16];                                   // 16 t-tiles of C/D (128 VGPRs)
#pragma unroll
    for (int t = 0; t < 16; ++t) acc[t] = (v8f){bv, bv, bv, bv, bv, bv, bv, bv};

    const float* hidA = hidden + ((size_t)b * (TT + 1) + 1) * HH;  // skip [CLS]

    // prologue: async-stage chunk 0
    stage_chunk_async(hidA, 0, ldsA0, tid);
    wait_async_zero();
    __syncthreads();

    for (int kc = 0; kc < HH; kc += KC) {
        float* cur = ((kc / KC) & 1) ? ldsA1 : ldsA0;
        float* nxt = ((kc / KC) & 1) ? ldsA0 : ldsA1;

        // kick off next chunk's HBM->LDS copies; they overlap the WMMAs below
        if (kc + KC < HH) stage_chunk_async(hidA, kc + KC, nxt, tid);

        for (int k8 = 0; k8 < KC / 4; ++k8) {
            // B fragment (4x16): VGPR0 = row K=kk, VGPR1 = row K=kk+1 (lanes 16-31 -> +2)
            const int kk = k8 * 4 + hi2;
            const float* wp = W + (size_t)(kc + kk) * SS + scc;
            v2f bf;
            bf.x = wp[0];
            bf.y = wp[SS];
#pragma unroll
            for (int t = 0; t < 16; ++t) {
                // A fragment (16x4): lane m holds (K=kk, K=kk+1) as two consecutive floats
                v2f af = *(const v2f*)&cur[(t * 16 + m) * PA + kk];
                acc[t] = __builtin_amdgcn_wmma_f32_16x16x4_f32(
                    false, af, false, bf, (short)0, acc[t], false, false);
            }
        }

        wait_async_zero();   // my async copies into nxt are done
        __syncthreads();     // everyone's copies + reads done -> nxt ready, cur reusable
    }

    // spill accumulators to the LDS logits buffer: row = t*16 + r + hi*8, col = sc
    if (sc < SS) {
#pragma unroll
        for (int t = 0; t < 16; ++t) {
#pragma unroll
            for (int r = 0; r < 8; ++r) {
                ldsL[(t * 16 + r + hi * 8) * PL + sc] = acc[t][r];
            }
        }
    }
    __syncthreads();

    // ---- phase 2: ragged segment mean + transpose, streamed out as [S, T] ----
    float* outb = out + (size_t)b * SS * TT;
    for (int idx = tid; idx < SS * TT; idx += NTHREADS) {
        int s = idx >> 8;          // / TT
        int w = idx & 255;         // % TT  (consecutive lanes -> consecutive w: coalesced)
        int c = ldsCnt[w];
        float r = 0.0f;
        if (c > 0) {
            int st = ldsStart[w];
            float sum = 0.0f;
            for (int i = 0; i < c; ++i) sum += ldsL[(st + i) * PL + s];
            r = sum / (float)c;
        }
        outb[(size_t)s * TT + w] = r;
    }
}

extern "C" void kernel_launch(void* const* d_in, const int* in_sizes, int n_in,
                              void* d_out, int out_size, void* d_ws, size_t ws_size,
                              hipStream_t stream) {
    const float* hidden = (const float*)d_in[0];   // [256,257,768] f32
    const float* W      = (const float*)d_in[1];   // [768,130]     f32
    const float* bias   = (const float*)d_in[2];   // [130]         f32
    const int*   seg    = (const int*)  d_in[3];   // [256,256]     i32
    float*       out    = (float*)d_out;           // [256,130,256] f32

    (void)in_sizes; (void)n_in; (void)out_size; (void)d_ws; (void)ws_size;

    hipFuncSetAttribute((const void*)slotmerge_kernel,
                        hipFuncAttributeMaxDynamicSharedMemorySize, LDS_BYTES);
    slotmerge_kernel<<<dim3(BB), dim3(NTHREADS), LDS_BYTES, stream>>>(
        hidden, W, bias, seg, out);
}